// ChannelRoll_23364622090325
// MI455X (gfx1250) — compile-verified
//
#include <hip/hip_runtime.h>
#include <hip/hip_bf16.h>

// ChannelRoll: out[row, c] = x[row, (c + m(row)) % 256], row in [0, B*H*W)
// Pure bandwidth kernel (~205 MB total). Strategy: wave32-per-row, stage the
// 1KB row into LDS with CDNA5 async global->LDS b128 DMA (ASYNCcnt), rotate
// via ds_load_b32 gather, write back with aligned non-temporal b128 stores.

typedef float v4f __attribute__((ext_vector_type(4)));

#define C_CHANNELS 256
#define WAVES_PER_BLOCK 8

__global__ void __launch_bounds__(256) ChannelRoll_kernel(
    const float* __restrict__ x,
    const int*   __restrict__ shift,   // int32 per harness dtype rules
    float*       __restrict__ out,
    int rows)
{
    __shared__ float smem[WAVES_PER_BLOCK * C_CHANNELS];   // 8 KB

    const int lane = threadIdx.x & 31;
    const int wave = threadIdx.x >> 5;
    const int row  = blockIdx.x * WAVES_PER_BLOCK + wave;  // wave-uniform
    if (row >= rows) return;                               // uniform branch

    const float* src = x   + (size_t)row * C_CHANNELS;
    float*       dst = out + (size_t)row * C_CHANNELS;

    // LDS byte address of this wave's slice: low 32 bits of the flat address
    // of a __shared__ object are the LDS offset (ISA 10.2 aperture mapping).
    unsigned lds0 = (unsigned)(size_t)(&smem[wave * C_CHANNELS]) + (unsigned)(lane * 16);
    unsigned long long g0 = (unsigned long long)(size_t)src + (unsigned long long)(lane * 16);

    // Two async 128-bit DMA loads per lane: 2 x 512B per wave = full 1KB row.
    // INST_OFFSET applies to both the LDS and global side (ISA 15.18.3 op 98).
    asm volatile("global_load_async_to_lds_b128 %0, %1, off"
                 :: "v"(lds0), "v"(g0) : "memory");
    asm volatile("global_load_async_to_lds_b128 %0, %1, off offset:512"
                 :: "v"(lds0), "v"(g0) : "memory");

    // Fetch the per-row shift while the DMA is in flight (LOADcnt, not ASYNCcnt).
    int m = shift[row] & (C_CHANNELS - 1);
    m = __builtin_amdgcn_readfirstlane(m);   // wave-uniform -> SGPR

    // Wait only on the async counter (CDNA5 split wait counters).
#if __has_builtin(__builtin_amdgcn_s_wait_asynccnt)
    __builtin_amdgcn_s_wait_asynccnt(0);
#else
    asm volatile("s_wait_asynccnt 0" ::: "memory");
#endif
    // Same wave issued the DMA, reads its own slice: no workgroup barrier needed.

    const float* wsm = &smem[wave * C_CHANNELS];
    #pragma unroll
    for (int k = 0; k < 2; ++k) {
        const int c = k * 128 + lane * 4;            // lanes cover 128 floats/iter
        v4f o;
        o.x = wsm[(c + 0 + m) & (C_CHANNELS - 1)];   // ds_load_b32 gather
        o.y = wsm[(c + 1 + m) & (C_CHANNELS - 1)];
        o.z = wsm[(c + 2 + m) & (C_CHANNELS - 1)];
        o.w = wsm[(c + 3 + m) & (C_CHANNELS - 1)];
        // Output is write-once/never-read: non-temporal b128 store keeps the
        // stream out of the 192MB L2 working set.
        __builtin_nontemporal_store(o, (v4f*)(dst + c));
    }
}

extern "C" void kernel_launch(void* const* d_in, const int* in_sizes, int n_in,
                              void* d_out, int out_size, void* d_ws, size_t ws_size,
                              hipStream_t stream) {
    (void)n_in; (void)out_size; (void)d_ws; (void)ws_size;
    const float* x     = (const float*)d_in[0];
    const int*   shift = (const int*)d_in[1];
    float*       out   = (float*)d_out;

    const int rows = in_sizes[1];   // B*H*W (shift_map element count)
    const int blocks = (rows + WAVES_PER_BLOCK - 1) / WAVES_PER_BLOCK;

    ChannelRoll_kernel<<<blocks, 256, 0, stream>>>(x, shift, out, rows);
}